// Loss_OHEM_10402410791783
// MI455X (gfx1250) — compile-verified
//
#include <hip/hip_runtime.h>
#include <hip/hip_bf16.h>

typedef __attribute__((ext_vector_type(2))) float v2f;
typedef __attribute__((ext_vector_type(8))) float v8f;

#define THREADS 256
#define BLOCKS  960

// ---------------- workspace layout ----------------
struct Sel {
    float pos_cnt;   // exact integer-valued float
    float pos_sum;
    float tot_sum;   // pos + neg loss sum
    float sum_sel;   // partial selected sum during radix select
    unsigned active; // refinement needed
    unsigned prefix; // accumulated bit prefix of threshold
    unsigned r;      // remaining count needed inside threshold bin
    unsigned done;   // selection finished
    float topk;      // final top-k sum
    unsigned pad[3];
};

struct WS {
    Sel sel[2];
    unsigned h0c[2][1024];  float h0s[2][1024];   // level 0: bits [30:21]
    unsigned h1c[2][2048];  float h1s[2][2048];   // level 1: bits [20:10]
    unsigned h2c[2][1024];  float h2s[2][1024];   // level 2: bits [9:0]
};

// ---------------- helpers ----------------
__device__ __forceinline__ void elem(float gv, float pv, float cv, float& l, bool& pos) {
    float d = gv - pv;
    l   = d * d * cv;                // two v_mul_f32; bit-identical across kernels
    pos = gv > 0.1f;
}

__device__ __forceinline__ float wave_sum16(float s) {
    // D[16x16 f32]: lanes 0-15 hold rows 0-7, lanes 16-31 rows 8-15 (replicated over N)
    unsigned a = __builtin_amdgcn_readlane(__float_as_uint(s), 0);
    unsigned b = __builtin_amdgcn_readlane(__float_as_uint(s), 16);
    return __uint_as_float(a) + __uint_as_float(b);
}

__device__ __forceinline__ v8f wmma_acc(v2f a, v8f c) {
    const v2f ones = {1.0f, 1.0f};
    // D = A(16x4) x Ones(4x16) + C  -> row sums accumulate 64 values/instruction
    return __builtin_amdgcn_wmma_f32_16x16x4_f32(false, a, false, ones,
                                                 (short)0, c, false, false);
}

// Process one float4 from each stream. `hist_valid` gates histogram updates only;
// WMMA calls sit at uniform control flow (EXEC all-1s).
__device__ __forceinline__ void quad_body(bool hist_valid, const float4& cf,
                                          const float4 g[2], const float4 q[2],
                                          v8f accCnt[2], v8f accPos[2], v8f accTot[2],
                                          unsigned hc[2][1024], float hs[2][1024]) {
    #pragma unroll
    for (int j = 0; j < 2; ++j) {
        float l[4]; bool pos[4];
        elem(g[j].x, q[j].x, cf.x, l[0], pos[0]);
        elem(g[j].y, q[j].y, cf.y, l[1], pos[1]);
        elem(g[j].z, q[j].z, cf.z, l[2], pos[2]);
        elem(g[j].w, q[j].w, cf.w, l[3], pos[3]);
        #pragma unroll
        for (int e = 0; e < 4; ++e) {
            if (hist_valid && !pos[e]) {             // negative -> histogram
                unsigned u = __float_as_uint(l[e]);  // l >= 0: bits monotonic
                unsigned b = u >> 21;                // 10 bits (bit31 == 0)
                atomicAdd(&hc[j][b], 1u);            // ds_add_u32
                atomicAdd(&hs[j][b], l[e]);          // ds_add_f32
            }
        }
        float pf[4], lp[4];
        #pragma unroll
        for (int e = 0; e < 4; ++e) {
            pf[e] = pos[e] ? 1.0f : 0.0f;
            lp[e] = pos[e] ? l[e] : 0.0f;
        }
        v2f A;
        A = (v2f){pf[0] + pf[2], pf[1] + pf[3]};
        accCnt[j] = wmma_acc(A, accCnt[j]);
        A = (v2f){lp[0] + lp[2], lp[1] + lp[3]};
        accPos[j] = wmma_acc(A, accPos[j]);
        A = (v2f){l[0] + l[2], l[1] + l[3]};
        accTot[j] = wmma_acc(A, accTot[j]);
    }
}

// ---------------- zero workspace ----------------
__global__ void zero_ws(unsigned* __restrict__ w, int nwords) {
    int i = blockIdx.x * blockDim.x + threadIdx.x;
    int stride = gridDim.x * blockDim.x;
    for (; i < nwords; i += stride) w[i] = 0u;
}

// ---------------- main fused pass ----------------
__global__ __launch_bounds__(THREADS)
void main_pass(const float* __restrict__ gr, const float* __restrict__ pr,
               const float* __restrict__ ga, const float* __restrict__ pa,
               const float* __restrict__ cm, int n, WS* __restrict__ ws) {
    __shared__ unsigned hc[2][1024];
    __shared__ float    hs[2][1024];
    for (int i = threadIdx.x; i < 2048; i += THREADS) {
        (&hc[0][0])[i] = 0u; (&hs[0][0])[i] = 0.0f;
    }
    __syncthreads();

    v8f accCnt[2] = {{}, {}}, accPos[2] = {{}, {}}, accTot[2] = {{}, {}};

    const int nv     = n >> 2;
    const int stride = (int)(gridDim.x * blockDim.x);
    const int tid0   = (int)(blockIdx.x * blockDim.x + threadIdx.x);
    const int full   = nv / stride;          // iterations where every lane is valid

    // ---- steady state: no guards, no zero-fill, dense b128 clauses ----
    for (int it = 0; it < full; ++it) {
        int idx = tid0 + it * stride;
        float4 cf   = ((const float4*)cm)[idx];
        float4 g[2] = {((const float4*)gr)[idx], ((const float4*)ga)[idx]};
        float4 q[2] = {((const float4*)pr)[idx], ((const float4*)pa)[idx]};
        quad_body(true, cf, g, q, accCnt, accPos, accTot, hc, hs);
    }
    // ---- single guarded epilogue iteration (loads gated; WMMA uniform) ----
    {
        int  idx   = tid0 + full * stride;
        bool valid = idx < nv;
        float4 cf = {0, 0, 0, 0};
        float4 g[2] = {{0,0,0,0},{0,0,0,0}};
        float4 q[2] = {{0,0,0,0},{0,0,0,0}};
        if (valid) {
            cf   = ((const float4*)cm)[idx];
            g[0] = ((const float4*)gr)[idx];  q[0] = ((const float4*)pr)[idx];
            g[1] = ((const float4*)ga)[idx];  q[1] = ((const float4*)pa)[idx];
        }
        quad_body(valid, cf, g, q, accCnt, accPos, accTot, hc, hs);
    }

    // scalar tail (n % 4), handled by block 0 without WMMA
    if (blockIdx.x == 0) {
        int tail = n & 3;
        if ((int)threadIdx.x < tail) {
            int i = (n & ~3) + (int)threadIdx.x;
            float cv = cm[i];
            const float* gp[2] = {gr, ga};
            const float* pp[2] = {pr, pa};
            for (int j = 0; j < 2; ++j) {
                float l; bool pos; elem(gp[j][i], pp[j][i], cv, l, pos);
                atomicAdd(&ws->sel[j].pos_cnt, pos ? 1.0f : 0.0f);
                atomicAdd(&ws->sel[j].pos_sum, pos ? l : 0.0f);
                atomicAdd(&ws->sel[j].tot_sum, l);
                if (!pos) {
                    unsigned u = __float_as_uint(l);
                    atomicAdd(&ws->h0c[j][u >> 21], 1u);
                    atomicAdd(&ws->h0s[j][u >> 21], l);
                }
            }
        }
    }

    __syncthreads();
    // flush block histogram
    for (int i = threadIdx.x; i < 1024; i += THREADS) {
        #pragma unroll
        for (int j = 0; j < 2; ++j) {
            unsigned cv = hc[j][i];
            if (cv) {
                atomicAdd(&ws->h0c[j][i], cv);
                atomicAdd(&ws->h0s[j][i], hs[j][i]);
            }
        }
    }
    // per-wave reduce of WMMA accumulators, one atomic per wave
    int lane = threadIdx.x & 31;
    #pragma unroll
    for (int j = 0; j < 2; ++j) {
        float sc = 0.0f, sp = 0.0f, st = 0.0f;
        #pragma unroll
        for (int v = 0; v < 8; ++v) { sc += accCnt[j][v]; sp += accPos[j][v]; st += accTot[j][v]; }
        float tc = wave_sum16(sc), tp = wave_sum16(sp), tt = wave_sum16(st);
        if (lane == 0) {
            atomicAdd(&ws->sel[j].pos_cnt, tc);
            atomicAdd(&ws->sel[j].pos_sum, tp);
            atomicAdd(&ws->sel[j].tot_sum, tt);
        }
    }
}

// ---------------- finalize level 0 ----------------
__global__ __launch_bounds__(THREADS)
void finalize0(WS* __restrict__ ws, int n) {
    __shared__ unsigned c[1024];
    __shared__ float    s[1024];
    for (int j = 0; j < 2; ++j) {
        for (int i = threadIdx.x; i < 1024; i += THREADS) { c[i] = ws->h0c[j][i]; s[i] = ws->h0s[j][i]; }
        __syncthreads();
        if (threadIdx.x == 0) {
            Sel& S = ws->sel[j];
            float pos     = S.pos_cnt;
            float negc    = (float)n - pos;             // exact (both < 2^24)
            float neg_num = fminf(negc, 3.0f * pos);
            int   k       = (int)neg_num;               // trunc, matches .astype(int32)
            float neg_sum = S.tot_sum - S.pos_sum;
            if (k <= 0)                 { S.done = 1; S.topk = 0.0f; }
            else if ((float)k >= negc)  { S.done = 1; S.topk = neg_sum; }  // fast path
            else {
                unsigned r = (unsigned)k; float acc = 0.0f;
                for (int b = 1023; b >= 0; --b) {
                    unsigned cb = c[b]; if (!cb) continue;
                    if (r > cb)       { acc += s[b]; r -= cb; }
                    else if (r == cb) { acc += s[b]; S.done = 1; S.topk = acc; break; }
                    else { S.active = 1; S.prefix = (unsigned)b; S.r = r; S.sum_sel = acc; break; }
                }
            }
        }
        __syncthreads();
    }
}

// ---------------- refinement passes (flag-gated; no-op on fast path) --------
__global__ __launch_bounds__(THREADS)
void refine_pass(const float* __restrict__ gr, const float* __restrict__ pr,
                 const float* __restrict__ ga, const float* __restrict__ pa,
                 const float* __restrict__ cm, int n, WS* __restrict__ ws, int level) {
    unsigned a0 = (ws->sel[0].active && !ws->sel[0].done) ? 1u : 0u;
    unsigned a1 = (ws->sel[1].active && !ws->sel[1].done) ? 1u : 0u;
    if (!(a0 | a1)) return;                 // expected path: entire grid exits

    __shared__ unsigned hc[2][2048];
    __shared__ float    hs[2][2048];
    const int nb = (level == 1) ? 2048 : 1024;
    for (int i = threadIdx.x; i < 2 * 2048; i += THREADS) {
        (&hc[0][0])[i] = 0u; (&hs[0][0])[i] = 0.0f;
    }
    __syncthreads();

    unsigned pre[2] = {ws->sel[0].prefix, ws->sel[1].prefix};
    unsigned act[2] = {a0, a1};

    const int nv     = n >> 2;
    const int stride = (int)(gridDim.x * blockDim.x);
    for (int idx = (int)(blockIdx.x * blockDim.x + threadIdx.x); idx < nv; idx += stride) {
        float4 cf   = ((const float4*)cm)[idx];
        float4 g[2] = {((const float4*)gr)[idx], ((const float4*)ga)[idx]};
        float4 q[2] = {((const float4*)pr)[idx], ((const float4*)pa)[idx]};
        #pragma unroll
        for (int j = 0; j < 2; ++j) {
            if (!act[j]) continue;
            float l[4]; bool pos[4];
            elem(g[j].x, q[j].x, cf.x, l[0], pos[0]);
            elem(g[j].y, q[j].y, cf.y, l[1], pos[1]);
            elem(g[j].z, q[j].z, cf.z, l[2], pos[2]);
            elem(g[j].w, q[j].w, cf.w, l[3], pos[3]);
            #pragma unroll
            for (int e = 0; e < 4; ++e) {
                if (pos[e]) continue;
                unsigned u = __float_as_uint(l[e]);
                if (level == 1) {
                    if ((u >> 21) == pre[j]) {
                        unsigned b = (u >> 10) & 0x7FFu;
                        atomicAdd(&hc[j][b], 1u); atomicAdd(&hs[j][b], l[e]);
                    }
                } else {
                    if ((u >> 10) == pre[j]) {
                        unsigned b = u & 0x3FFu;
                        atomicAdd(&hc[j][b], 1u); atomicAdd(&hs[j][b], l[e]);
                    }
                }
            }
        }
    }
    // scalar tail
    if (blockIdx.x == 0) {
        int tail = n & 3;
        if ((int)threadIdx.x < tail) {
            int i = (n & ~3) + (int)threadIdx.x;
            float cv = cm[i];
            const float* gp[2] = {gr, ga};
            const float* pp[2] = {pr, pa};
            for (int j = 0; j < 2; ++j) {
                if (!act[j]) continue;
                float l; bool pos; elem(gp[j][i], pp[j][i], cv, l, pos);
                if (pos) continue;
                unsigned u = __float_as_uint(l);
                if (level == 1) { if ((u >> 21) == pre[j]) { atomicAdd(&hc[j][(u>>10)&0x7FFu],1u); atomicAdd(&hs[j][(u>>10)&0x7FFu],l);} }
                else            { if ((u >> 10) == pre[j]) { atomicAdd(&hc[j][u&0x3FFu],1u);       atomicAdd(&hs[j][u&0x3FFu],l);} }
            }
        }
    }
    __syncthreads();
    for (int i = threadIdx.x; i < nb; i += THREADS) {
        #pragma unroll
        for (int j = 0; j < 2; ++j) {
            unsigned cv = hc[j][i];
            if (cv) {
                unsigned* gc = (level == 1) ? &ws->h1c[j][0] : &ws->h2c[j][0];
                float*    gs = (level == 1) ? &ws->h1s[j][0] : &ws->h2s[j][0];
                atomicAdd(&gc[i], cv); atomicAdd(&gs[i], hs[j][i]);
            }
        }
    }
}

// ---------------- finalize level 1 ----------------
__global__ __launch_bounds__(THREADS)
void finalize1(WS* __restrict__ ws) {
    __shared__ unsigned c[2048];
    __shared__ float    s[2048];
    for (int j = 0; j < 2; ++j) {
        for (int i = threadIdx.x; i < 2048; i += THREADS) { c[i] = ws->h1c[j][i]; s[i] = ws->h1s[j][i]; }
        __syncthreads();
        if (threadIdx.x == 0) {
            Sel& S = ws->sel[j];
            if (S.active && !S.done) {
                unsigned r = S.r; float acc = S.sum_sel;
                for (int b = 2047; b >= 0; --b) {
                    unsigned cb = c[b]; if (!cb) continue;
                    if (r > cb)       { acc += s[b]; r -= cb; }
                    else if (r == cb) { acc += s[b]; S.done = 1; S.topk = acc; break; }
                    else { S.prefix = (S.prefix << 11) | (unsigned)b; S.r = r; S.sum_sel = acc; break; }
                }
            }
        }
        __syncthreads();
    }
}

// ---------------- finalize level 2 + output ----------------
__global__ __launch_bounds__(THREADS)
void finalize2(WS* __restrict__ ws, int n, float* __restrict__ out) {
    __shared__ unsigned c[1024];
    __shared__ float    s[1024];
    for (int j = 0; j < 2; ++j) {
        for (int i = threadIdx.x; i < 1024; i += THREADS) { c[i] = ws->h2c[j][i]; s[i] = ws->h2s[j][i]; }
        __syncthreads();
        if (threadIdx.x == 0) {
            Sel& S = ws->sel[j];
            if (S.active && !S.done) {
                unsigned r = S.r; float acc = S.sum_sel;
                for (int b = 1023; b >= 0; --b) {
                    unsigned cb = c[b]; if (!cb) continue;
                    if (r > cb) { acc += s[b]; r -= cb; }
                    else {      // bins here are single exact float values
                        float V = __uint_as_float((S.prefix << 10) | (unsigned)b);
                        S.topk = acc + (float)r * V; S.done = 1; break;
                    }
                }
                if (!S.done) { S.topk = acc; S.done = 1; }   // safety
            }
        }
        __syncthreads();
    }
    if (threadIdx.x == 0) {
        float total = 0.0f;
        for (int j = 0; j < 2; ++j) {
            Sel& S = ws->sel[j];
            float pos     = S.pos_cnt;
            float neg_num = fminf((float)n - pos, 3.0f * pos);
            total += (S.topk + S.pos_sum) / (neg_num + pos);
        }
        out[0] = total;
    }
}

// ---------------- host launch ----------------
extern "C" void kernel_launch(void* const* d_in, const int* in_sizes, int n_in,
                              void* d_out, int out_size, void* d_ws, size_t ws_size,
                              hipStream_t stream) {
    const float* gr = (const float*)d_in[0];
    const float* pr = (const float*)d_in[1];
    const float* ga = (const float*)d_in[2];
    const float* pa = (const float*)d_in[3];
    const float* cm = (const float*)d_in[4];
    int n  = in_sizes[0];
    WS* ws = (WS*)d_ws;

    zero_ws<<<64, THREADS, 0, stream>>>((unsigned*)d_ws, (int)(sizeof(WS) / 4));
    main_pass<<<BLOCKS, THREADS, 0, stream>>>(gr, pr, ga, pa, cm, n, ws);
    finalize0<<<1, THREADS, 0, stream>>>(ws, n);
    refine_pass<<<BLOCKS, THREADS, 0, stream>>>(gr, pr, ga, pa, cm, n, ws, 1);
    finalize1<<<1, THREADS, 0, stream>>>(ws);
    refine_pass<<<BLOCKS, THREADS, 0, stream>>>(gr, pr, ga, pa, cm, n, ws, 2);
    finalize2<<<1, THREADS, 0, stream>>>(ws, n, (float*)d_out);
}